// GraphFeatureSelector_2405181686012
// MI455X (gfx1250) — compile-verified
//
#include <hip/hip_runtime.h>
#include <hip/hip_bf16.h>
#include <math.h>

// Problem constants (from reference setup_inputs)
#define BB   8
#define ORD  256
#define GG   (BB * ORD)          // 2048 graphs
#define NN   51                  // nodes
#define NP   64                  // padded node count (4 x 16 tiles)
#define HH   16                  // hidden
#define KSEL 10                  // top-k
#define PP   32                  // projection dim
#define KH   (KSEL * HH)         // 160

typedef float v2f __attribute__((ext_vector_type(2)));
typedef float v8f __attribute__((ext_vector_type(8)));

__device__ __forceinline__ float wave_max32(float v) {
    for (int o = 16; o > 0; o >>= 1) v = fmaxf(v, __shfl_xor(v, o, 32));
    return v;
}
__device__ __forceinline__ float wave_sum32(float v) {
    for (int o = 16; o > 0; o >>= 1) v += __shfl_xor(v, o, 32);
    return v;
}

// One workgroup per graph g. 256 threads = 8 waves (wave32).
__global__ __launch_bounds__(256)
void graph_kernel(const float* __restrict__ x,        // [G, N]
                  const float* __restrict__ u_noise,  // [G, N, N]
                  const float* __restrict__ phi,      // [N, N]
                  const float* __restrict__ Wcat_w,   // [2, H]
                  const float* __restrict__ Wcat_b,   // [H]
                  const float* __restrict__ a_w,      // [H]
                  const float* __restrict__ Wnode_w,  // [1, H]
                  const float* __restrict__ Wnode_b,  // [H]
                  float* __restrict__ out_topk,       // [G, K] (as float)
                  float* __restrict__ out_A,          // [G, N, N]
                  float* __restrict__ sel)            // ws: [G, K*H]
{
    const int g    = blockIdx.x;
    const int tid  = threadIdx.x;
    const int lane = tid & 31;
    const int wave = tid >> 5;

    __shared__ float xg[NP];
    __shared__ float aws[HH];
    __shared__ float c1[NP][HH];
    __shared__ float c2[NP][HH];
    __shared__ float alpha[NP][NP];   // attention, padded with zeros
    __shared__ float npj[NP][HH];     // node_proj, padded with zeros
    __shared__ float emb[NP][HH];     // node_emb
    __shared__ float norms[NP];
    __shared__ int   tki[KSEL];

    // Stage x row + attention vector; zero alpha padding region.
    if (tid < NP) xg[tid] = (tid < NN) ? x[(size_t)g * NN + tid] : 0.0f;
    if (tid < HH) aws[tid] = a_w[tid];
    for (int i = tid; i < NP * NP; i += 256) (&alpha[0][0])[i] = 0.0f;
    __syncthreads();

    // c1[i][h] = x_i * W1[h]; c2[j][h] = x_j * W2[h] + b[h]; node_proj.
    for (int i = tid; i < NP * HH; i += 256) {
        const int r = i >> 4, h = i & 15;
        c1[r][h]  = xg[r] * Wcat_w[h];
        c2[r][h]  = xg[r] * Wcat_w[HH + h] + Wcat_b[h];
        npj[r][h] = (r < NN) ? (xg[r] * Wnode_w[h] + Wnode_b[h]) : 0.0f;
    }
    __syncthreads();

    // Per-row: Gumbel-softmax A, then MLP scores + log(A+eps), softmax -> alpha.
    const float NEG = -1e30f;
    for (int i = wave; i < NN; i += 8) {
        const int j0 = lane, j1 = lane + 32;
        float l0 = NEG, l1 = NEG;
        if (j0 < NN) {
            float u  = u_noise[((size_t)g * NN + i) * NN + j0];
            float gu = -logf(-logf(u + 1e-9f) + 1e-9f);
            l0 = (phi[i * NN + j0] + gu) * 2.0f;   // / TEMP (0.5)
        }
        if (j1 < NN) {
            float u  = u_noise[((size_t)g * NN + i) * NN + j1];
            float gu = -logf(-logf(u + 1e-9f) + 1e-9f);
            l1 = (phi[i * NN + j1] + gu) * 2.0f;
        }
        float m  = wave_max32(fmaxf(l0, l1));
        float e0 = (j0 < NN) ? expf(l0 - m) : 0.0f;
        float e1 = (j1 < NN) ? expf(l1 - m) : 0.0f;
        float s  = wave_sum32(e0 + e1);
        float A0 = e0 / s, A1 = e1 / s;
        if (j0 < NN) out_A[((size_t)g * NN + i) * NN + j0] = A0;
        if (j1 < NN) out_A[((size_t)g * NN + i) * NN + j1] = A1;

        float s0 = NEG, s1 = NEG;
        if (j0 < NN) {
            float acc = 0.0f;
            #pragma unroll
            for (int h = 0; h < HH; ++h) {
                float v = c1[i][h] + c2[j0][h];
                v = (v > 0.0f) ? v : 0.2f * v;     // leaky_relu(0.2)
                acc += aws[h] * v;
            }
            s0 = acc + logf(A0 + 1e-8f);
        }
        if (j1 < NN) {
            float acc = 0.0f;
            #pragma unroll
            for (int h = 0; h < HH; ++h) {
                float v = c1[i][h] + c2[j1][h];
                v = (v > 0.0f) ? v : 0.2f * v;
                acc += aws[h] * v;
            }
            s1 = acc + logf(A1 + 1e-8f);
        }
        m  = wave_max32(fmaxf(s0, s1));
        e0 = (j0 < NN) ? expf(s0 - m) : 0.0f;
        e1 = (j1 < NN) ? expf(s1 - m) : 0.0f;
        s  = wave_sum32(e0 + e1);
        if (j0 < NN) alpha[i][j0] = e0 / s;
        if (j1 < NN) alpha[i][j1] = e1 / s;
    }
    __syncthreads();

    // node_emb[64x16] = relu(alpha[64x64] @ node_proj[64x16]) via f32 WMMA.
    // Waves 0..3 each own one 16-row M-tile; 16 K-steps of 16x16x4.
    if (wave < 4) {
        const int mt   = wave;
        const int half = lane >> 4;
        const int l16  = lane & 15;
        const int row  = mt * 16 + l16;
        v8f acc = {};
        #pragma unroll
        for (int kb = 0; kb < 16; ++kb) {
            const int k0 = kb * 4 + 2 * half;
            v2f a, b;
            a.x = alpha[row][k0];
            a.y = alpha[row][k0 + 1];
            b.x = npj[k0][l16];
            b.y = npj[k0 + 1][l16];
            acc = __builtin_amdgcn_wmma_f32_16x16x4_f32(
                false, a, false, b, (short)0, acc, false, false);
        }
        #pragma unroll
        for (int r = 0; r < 8; ++r) {
            emb[mt * 16 + r + 8 * half][l16] = fmaxf(acc[r], 0.0f);
        }
    }
    __syncthreads();

    // Row norms; pad rows get -1 so they never enter top-k.
    if (tid < NP) {
        if (tid < NN) {
            float s = 0.0f;
            #pragma unroll
            for (int h = 0; h < HH; ++h) s += emb[tid][h] * emb[tid][h];
            norms[tid] = sqrtf(s);
        } else {
            norms[tid] = -1.0f;
        }
    }
    __syncthreads();

    // Serial top-K (strict '>' keeps earliest index on ties, matching lax.top_k).
    if (tid == 0) {
        bool taken[NN];
        #pragma unroll
        for (int i = 0; i < NN; ++i) taken[i] = false;
        for (int k = 0; k < KSEL; ++k) {
            float best = -2.0f; int bi = 0;
            for (int i = 0; i < NN; ++i)
                if (!taken[i] && norms[i] > best) { best = norms[i]; bi = i; }
            taken[bi] = true;
            tki[k] = bi;
        }
    }
    __syncthreads();

    if (tid < KSEL) out_topk[(size_t)g * KSEL + tid] = (float)tki[tid];
    if (tid < KH) {
        const int k = tid >> 4, h = tid & 15;
        sel[(size_t)g * KH + tid] = emb[tki[k]][h];
    }
}

// projected[2048x32] = sel[2048x160] @ proj_w[160x32] + proj_b.
// One wave per 16x16 output tile; 40 K-steps of f32 16x16x4 WMMA.
__global__ __launch_bounds__(32)
void proj_kernel(const float* __restrict__ sel,   // [G, 160]
                 const float* __restrict__ pw,    // [160, 32]
                 const float* __restrict__ pb,    // [32]
                 float* __restrict__ out)         // [G, 32]
{
    const int nt   = blockIdx.x & 1;
    const int mt   = blockIdx.x >> 1;
    const int lane = threadIdx.x & 31;
    const int half = lane >> 4;
    const int l16  = lane & 15;
    const int row  = mt * 16 + l16;
    const int col  = nt * 16 + l16;

    v8f acc = {};
    #pragma unroll 4
    for (int kb = 0; kb < 40; ++kb) {
        const int k0 = kb * 4 + 2 * half;
        v2f a, b;
        a.x = sel[(size_t)row * KH + k0];
        a.y = sel[(size_t)row * KH + k0 + 1];
        b.x = pw[(size_t)k0 * PP + col];
        b.y = pw[(size_t)(k0 + 1) * PP + col];
        acc = __builtin_amdgcn_wmma_f32_16x16x4_f32(
            false, a, false, b, (short)0, acc, false, false);
    }
    const float bias = pb[col];
    #pragma unroll
    for (int r = 0; r < 8; ++r) {
        const int orow = mt * 16 + r + 8 * half;
        out[(size_t)orow * PP + col] = acc[r] + bias;
    }
}

extern "C" void kernel_launch(void* const* d_in, const int* in_sizes, int n_in,
                              void* d_out, int out_size, void* d_ws, size_t ws_size,
                              hipStream_t stream) {
    const float* x        = (const float*)d_in[0];
    const float* u_noise  = (const float*)d_in[1];
    const float* phi      = (const float*)d_in[2];
    const float* Wcat_w   = (const float*)d_in[3];
    const float* Wcat_b   = (const float*)d_in[4];
    const float* a_w      = (const float*)d_in[5];
    const float* Wnode_w  = (const float*)d_in[6];
    const float* Wnode_b  = (const float*)d_in[7];
    const float* proj_w   = (const float*)d_in[8];
    const float* proj_b   = (const float*)d_in[9];

    float* out      = (float*)d_out;
    float* out_proj = out;                             // [G, 32]
    float* out_topk = out + (size_t)GG * PP;           // [G, 10]
    float* out_A    = out_topk + (size_t)GG * KSEL;    // [G, 51, 51]
    float* sel      = (float*)d_ws;                    // [G, 160]

    graph_kernel<<<GG, 256, 0, stream>>>(x, u_noise, phi, Wcat_w, Wcat_b, a_w,
                                         Wnode_w, Wnode_b, out_topk, out_A, sel);
    proj_kernel<<<(GG / 16) * (PP / 16), 32, 0, stream>>>(sel, proj_w, proj_b, out_proj);
}